// HRANGNN_87205015978178
// MI455X (gfx1250) — compile-verified
//
#include <hip/hip_runtime.h>
#include <hip/hip_bf16.h>
#include <math.h>

#define NN 4096
#define IN_F 256
#define H0 64
#define H1 64
#define H2 32
#define SLOPE 0.01f
#define CAP 512          // per-row nonzero list capacity (mean nnz ~ 41)

typedef float v2f __attribute__((ext_vector_type(2)));
typedef float v4f __attribute__((ext_vector_type(4)));
typedef float v8f __attribute__((ext_vector_type(8)));

__device__ __forceinline__ float leaky(float v) {
    return v >= 0.f ? v : SLOPE * v;
}

// ---------------------------------------------------------------------------
// WMMA fragment helpers (V_WMMA_F32_16X16X4_F32, ISA 7.12.2 layouts):
// A 16x4 f32: lanes 0-15 rows M=0..15 K{0,1}; lanes 16-31 rows M=0..15 K{2,3}.
// B 4x16 mirrors with N across lanes.  C/D: VGPR v holds M=v (+8 for upper half).
// ---------------------------------------------------------------------------
__device__ __forceinline__ void wmma_store_f32(float* __restrict__ C, int ldc,
                                               int m0, int n0, v8f c, float bias) {
    const int lane = threadIdx.x & 31;
    const int n    = n0 + (lane & 15);
    const int mb   = m0 + (lane >> 4) * 8;
    #pragma unroll
    for (int v = 0; v < 8; ++v)
        C[(size_t)(mb + v) * ldc + n] = c[v] + bias;
}

// ---------------------------------------------------------------------------
// K1: Wh[r] = x @ W_r.  x:[4096,256], W:[256,64].
// One wave -> full 16x64 row stripe: one A-frag feeds 4 WMMAs per k-step.
// 3*256 = 768 waves.
// ---------------------------------------------------------------------------
__global__ void k_gemm_wh3(const float* __restrict__ x,
                           const float* __restrict__ W1,
                           const float* __restrict__ W2,
                           const float* __restrict__ W3,
                           float* __restrict__ Wh) {
    const int wave = (int)((blockIdx.x * blockDim.x + threadIdx.x) >> 5);
    const int r    = wave >> 8;        // 0..2
    const int mt   = wave & 255;
    const float* W = (r == 0) ? W1 : (r == 1) ? W2 : W3;
    const int lane = threadIdx.x & 31;
    const int half = lane >> 4;
    const int l15  = lane & 15;
    const int arow = mt * 16 + l15;
    v8f c0 = {}, c1 = {}, c2 = {}, c3 = {};
    for (int k = 0; k < IN_F; k += 4) {
        const int ka = k + half * 2;
        v2f a, b0, b1, b2, b3;
        a.x  = x[(size_t)arow * IN_F + ka];
        a.y  = x[(size_t)arow * IN_F + ka + 1];
        b0.x = W[(size_t)ka * H0 + l15];       b0.y = W[(size_t)(ka + 1) * H0 + l15];
        b1.x = W[(size_t)ka * H0 + l15 + 16];  b1.y = W[(size_t)(ka + 1) * H0 + l15 + 16];
        b2.x = W[(size_t)ka * H0 + l15 + 32];  b2.y = W[(size_t)(ka + 1) * H0 + l15 + 32];
        b3.x = W[(size_t)ka * H0 + l15 + 48];  b3.y = W[(size_t)(ka + 1) * H0 + l15 + 48];
        c0 = __builtin_amdgcn_wmma_f32_16x16x4_f32(false, a, false, b0, (short)0, c0, false, false);
        c1 = __builtin_amdgcn_wmma_f32_16x16x4_f32(false, a, false, b1, (short)0, c1, false, false);
        c2 = __builtin_amdgcn_wmma_f32_16x16x4_f32(false, a, false, b2, (short)0, c2, false, false);
        c3 = __builtin_amdgcn_wmma_f32_16x16x4_f32(false, a, false, b3, (short)0, c3, false, false);
    }
    float* C = Wh + (size_t)r * NN * H0;
    wmma_store_f32(C, H0, mt * 16, 0,  c0, 0.f);
    wmma_store_f32(C, H0, mt * 16, 16, c1, 0.f);
    wmma_store_f32(C, H0, mt * 16, 32, c2, 0.f);
    wmma_store_f32(C, H0, mt * 16, 48, c3, 0.f);
}

// ---------------------------------------------------------------------------
// K2: sSrc = Wh[i] . a[:64], sDst = Wh[i] . a[64:]
// ---------------------------------------------------------------------------
__global__ void k_scores(const float* __restrict__ Wh,
                         const float* __restrict__ a1,
                         const float* __restrict__ a2,
                         const float* __restrict__ a3,
                         float* __restrict__ sSrc,
                         float* __restrict__ sDst) {
    const int idx = blockIdx.x * blockDim.x + threadIdx.x;   // 3*4096
    if (idx >= 3 * NN) return;
    const int r = idx >> 12;
    const float* a  = (r == 0) ? a1 : (r == 1) ? a2 : a3;
    const float* wh = Wh + (size_t)idx * H0;
    float s0 = 0.f, s1 = 0.f;
    #pragma unroll 8
    for (int f = 0; f < H0; ++f) {
        const float w = wh[f];
        s0 += w * a[f];
        s1 += w * a[H0 + f];
    }
    sSrc[idx] = s0;
    sDst[idx] = s1;
}

// ---------------------------------------------------------------------------
// K3: attention aggregation, block = (row, relation), 256 threads.
// Lane-interleaved b128 streaming of the 16KB adj row (NT hint for the two
// planes that are never re-read, keeping adj[relation] L2-resident for K6/K8),
// masked 2-phase softmax, nonzero compaction, feature-parallel gather.
// ---------------------------------------------------------------------------
__global__ void k_attn_aggregate(const float* __restrict__ adj,
                                 const int* __restrict__ relPtr,
                                 const float* __restrict__ Wh,
                                 const float* __restrict__ sSrc,
                                 const float* __restrict__ sDst,
                                 float* __restrict__ hattn) {
    const int i   = blockIdx.x;
    const int r   = blockIdx.y;
    const int tid = threadIdx.x;     // 256
    const float* arow = adj + ((size_t)r * NN + i) * NN;
    const float* WhR  = Wh + (size_t)r * NN * H0;
    const float  ssrc = sSrc[r * NN + i];
    const float* sd   = sDst + (size_t)r * NN;
    const bool   useNT = (r != relPtr[0]);

    __shared__ float red[256];
    __shared__ float acc[H0];
    __shared__ int   jidx[CAP];
    __shared__ float pval[CAP];
    __shared__ int   nnz;
    if (tid < H0) acc[tid] = 0.f;
    if (tid == 0) nnz = 0;

    // j(q,e) = 4*tid + 1024*q + e : each wave load covers contiguous 512B
    v4f av[4], sdv[4];
    const v4f* arow4 = (const v4f*)arow;
    const v4f* sd4   = (const v4f*)sd;
    if (useNT) {
        #pragma unroll
        for (int q = 0; q < 4; ++q)
            av[q] = __builtin_nontemporal_load(arow4 + tid + 256 * q);
    } else {
        #pragma unroll
        for (int q = 0; q < 4; ++q)
            av[q] = arow4[tid + 256 * q];
    }
    #pragma unroll
    for (int q = 0; q < 4; ++q)
        sdv[q] = sd4[tid + 256 * q];

    float sc[16];
    float lmax = -INFINITY;
    #pragma unroll
    for (int q = 0; q < 4; ++q)
        #pragma unroll
        for (int e = 0; e < 4; ++e) {
            const float v = leaky(ssrc + sdv[q][e]);
            sc[4 * q + e] = v;
            if (av[q][e] > 0.f) lmax = fmaxf(lmax, v);
        }
    red[tid] = lmax;
    __syncthreads();
    for (int s = 128; s > 0; s >>= 1) {
        if (tid < s) red[tid] = fmaxf(red[tid], red[tid + s]);
        __syncthreads();
    }
    const float rowmax = red[0];
    __syncthreads();

    float lsum = 0.f;
    #pragma unroll
    for (int q = 0; q < 4; ++q)
        #pragma unroll
        for (int e = 0; e < 4; ++e) {
            if (av[q][e] > 0.f) {
                const float pv = expf(sc[4 * q + e] - rowmax);
                lsum += pv;
                const int j   = 4 * tid + 1024 * q + e;
                const int pos = atomicAdd(&nnz, 1);
                if (pos < CAP) { jidx[pos] = j; pval[pos] = pv; }
                else {           // pathological overflow: direct accumulate
                    const float* whj = WhR + (size_t)j * H0;
                    for (int ff = 0; ff < H0; ++ff) {
                        const int f = (ff + tid) & (H0 - 1);
                        atomicAdd(&acc[f], pv * whj[f]);
                    }
                }
            }
        }
    red[tid] = lsum;
    __syncthreads();
    for (int s = 128; s > 0; s >>= 1) {
        if (tid < s) red[tid] += red[tid + s];
        __syncthreads();
    }
    const float denom = red[0];

    // feature-parallel gather over the compacted list (coalesced 256B/row)
    const int m     = (nnz < CAP) ? nnz : CAP;
    const int f     = tid & (H0 - 1);
    const int chunk = tid >> 6;                // 4 chunks
    float part = 0.f;
    for (int k = chunk; k < m; k += 4)
        part += pval[k] * WhR[(size_t)jidx[k] * H0 + f];
    atomicAdd(&acc[f], part);
    __syncthreads();
    if (tid < H0)
        hattn[((size_t)r * NN + i) * H0 + tid] = acc[tid] / denom;
}

// ---------------------------------------------------------------------------
// K4: h' = sigmoid((h1+h2+h3)/3)
// ---------------------------------------------------------------------------
__global__ void k_combine(const float* __restrict__ hattn, float* __restrict__ hp) {
    const int idx = blockIdx.x * blockDim.x + threadIdx.x;   // 4096*64
    const size_t S = (size_t)NN * H0;
    const float v = (hattn[idx] + hattn[S + idx] + hattn[2 * S + idx]) * (1.f / 3.f);
    hp[idx] = 1.f / (1.f + expf(-v));
}

// ---------------------------------------------------------------------------
// K5: support0 = h' @ Wg0  [4096,64]@[64,64].  One wave -> 16x64 stripe.
// ---------------------------------------------------------------------------
__global__ void k_gemm_support0(const float* __restrict__ hp,
                                const float* __restrict__ Wg0,
                                float* __restrict__ support0) {
    const int mt   = (int)((blockIdx.x * blockDim.x + threadIdx.x) >> 5);  // 256 waves
    const int lane = threadIdx.x & 31;
    const int half = lane >> 4;
    const int l15  = lane & 15;
    const int arow = mt * 16 + l15;
    v8f c0 = {}, c1 = {}, c2 = {}, c3 = {};
    for (int k = 0; k < H0; k += 4) {
        const int ka = k + half * 2;
        v2f a, b0, b1, b2, b3;
        a.x  = hp[(size_t)arow * H0 + ka];
        a.y  = hp[(size_t)arow * H0 + ka + 1];
        b0.x = Wg0[(size_t)ka * H1 + l15];       b0.y = Wg0[(size_t)(ka + 1) * H1 + l15];
        b1.x = Wg0[(size_t)ka * H1 + l15 + 16];  b1.y = Wg0[(size_t)(ka + 1) * H1 + l15 + 16];
        b2.x = Wg0[(size_t)ka * H1 + l15 + 32];  b2.y = Wg0[(size_t)(ka + 1) * H1 + l15 + 32];
        b3.x = Wg0[(size_t)ka * H1 + l15 + 48];  b3.y = Wg0[(size_t)(ka + 1) * H1 + l15 + 48];
        c0 = __builtin_amdgcn_wmma_f32_16x16x4_f32(false, a, false, b0, (short)0, c0, false, false);
        c1 = __builtin_amdgcn_wmma_f32_16x16x4_f32(false, a, false, b1, (short)0, c1, false, false);
        c2 = __builtin_amdgcn_wmma_f32_16x16x4_f32(false, a, false, b2, (short)0, c2, false, false);
        c3 = __builtin_amdgcn_wmma_f32_16x16x4_f32(false, a, false, b3, (short)0, c3, false, false);
    }
    wmma_store_f32(support0, H1, mt * 16, 0,  c0, 0.f);
    wmma_store_f32(support0, H1, mt * 16, 16, c1, 0.f);
    wmma_store_f32(support0, H1, mt * 16, 32, c2, 0.f);
    wmma_store_f32(support0, H1, mt * 16, 48, c3, 0.f);
}

// ---------------------------------------------------------------------------
// K6/K8: out[i] = leaky((1/deg_i) * sum_j adj_ij*support[j] + b) (+ resid).
// Same streaming/compaction structure as K3.  lastUse -> NT loads of adj.
// ---------------------------------------------------------------------------
__global__ void k_gnn_aggregate(const float* __restrict__ adj,
                                const int* __restrict__ relPtr,
                                const float* __restrict__ support,
                                const float* __restrict__ bias,
                                const float* __restrict__ resid,
                                float* __restrict__ out, int F, int lastUse) {
    const int i   = blockIdx.x;
    const int tid = threadIdx.x;
    const int rel = relPtr[0];
    const float* arow = adj + ((size_t)rel * NN + i) * NN;

    __shared__ float red[256];
    __shared__ float acc[H0];
    __shared__ int   jidx[CAP];
    __shared__ float pval[CAP];
    __shared__ int   nnz;
    if (tid < F) acc[tid] = 0.f;
    if (tid == 0) nnz = 0;

    v4f av[4];
    const v4f* arow4 = (const v4f*)arow;
    if (lastUse) {
        #pragma unroll
        for (int q = 0; q < 4; ++q)
            av[q] = __builtin_nontemporal_load(arow4 + tid + 256 * q);
    } else {
        #pragma unroll
        for (int q = 0; q < 4; ++q)
            av[q] = arow4[tid + 256 * q];
    }

    float lsum = 0.f;
    #pragma unroll
    for (int q = 0; q < 4; ++q)
        #pragma unroll
        for (int e = 0; e < 4; ++e) {
            const float a = av[q][e];
            lsum += a;
            if (a > 0.f) {
                const int j   = 4 * tid + 1024 * q + e;
                const int pos = atomicAdd(&nnz, 1);
                if (pos < CAP) { jidx[pos] = j; pval[pos] = a; }
                else {
                    const float* sj = support + (size_t)j * F;
                    for (int ff = 0; ff < F; ++ff) {
                        const int f = (ff + tid) & (F - 1);
                        atomicAdd(&acc[f], a * sj[f]);
                    }
                }
            }
        }
    red[tid] = lsum;
    __syncthreads();
    for (int s = 128; s > 0; s >>= 1) {
        if (tid < s) red[tid] += red[tid + s];
        __syncthreads();
    }
    const float deg = red[0];

    const int m     = (nnz < CAP) ? nnz : CAP;
    const int f     = tid & (F - 1);
    const int chunk = tid / F;
    const int step  = 256 / F;
    float part = 0.f;
    for (int k = chunk; k < m; k += step)
        part += pval[k] * support[(size_t)jidx[k] * F + f];
    atomicAdd(&acc[f], part);
    __syncthreads();
    if (tid < F) {
        const float dinv = deg > 0.f ? 1.f / deg : 0.f;
        float v = leaky(acc[tid] * dinv + bias[tid]);
        if (resid) v += resid[(size_t)i * F + tid];
        out[(size_t)i * F + tid] = v;
    }
}

// ---------------------------------------------------------------------------
// K7: support1 = out1 @ Wg1 and resid = out1 @ Wr^T + br, fused: one wave
// computes all four 16x16 tiles of a row stripe off a shared A-frag.
// ---------------------------------------------------------------------------
__global__ void k_gemm_layer2(const float* __restrict__ h,
                              const float* __restrict__ Wg1,
                              const float* __restrict__ Wr,
                              const float* __restrict__ br,
                              float* __restrict__ support1,
                              float* __restrict__ resid) {
    const int mt   = (int)((blockIdx.x * blockDim.x + threadIdx.x) >> 5);  // 256 waves
    const int lane = threadIdx.x & 31;
    const int half = lane >> 4;
    const int l15  = lane & 15;
    const int arow = mt * 16 + l15;
    v8f c0 = {}, c1 = {}, d0 = {}, d1 = {};
    for (int k = 0; k < H1; k += 4) {
        const int ka = k + half * 2;
        v2f a, b0, b1, e0, e1;
        a.x  = h[(size_t)arow * H1 + ka];
        a.y  = h[(size_t)arow * H1 + ka + 1];
        b0.x = Wg1[(size_t)ka * H2 + l15];       b0.y = Wg1[(size_t)(ka + 1) * H2 + l15];
        b1.x = Wg1[(size_t)ka * H2 + l15 + 16];  b1.y = Wg1[(size_t)(ka + 1) * H2 + l15 + 16];
        e0.x = Wr[(size_t)l15 * H1 + ka];        e0.y = Wr[(size_t)l15 * H1 + ka + 1];         // (Wr^T)[k][n]
        e1.x = Wr[(size_t)(l15 + 16) * H1 + ka]; e1.y = Wr[(size_t)(l15 + 16) * H1 + ka + 1];
        c0 = __builtin_amdgcn_wmma_f32_16x16x4_f32(false, a, false, b0, (short)0, c0, false, false);
        c1 = __builtin_amdgcn_wmma_f32_16x16x4_f32(false, a, false, b1, (short)0, c1, false, false);
        d0 = __builtin_amdgcn_wmma_f32_16x16x4_f32(false, a, false, e0, (short)0, d0, false, false);
        d1 = __builtin_amdgcn_wmma_f32_16x16x4_f32(false, a, false, e1, (short)0, d1, false, false);
    }
    wmma_store_f32(support1, H2, mt * 16, 0,  c0, 0.f);
    wmma_store_f32(support1, H2, mt * 16, 16, c1, 0.f);
    wmma_store_f32(resid,    H2, mt * 16, 0,  d0, br[l15]);
    wmma_store_f32(resid,    H2, mt * 16, 16, d1, br[l15 + 16]);
}

// ---------------------------------------------------------------------------
// launcher
// ---------------------------------------------------------------------------
extern "C" void kernel_launch(void* const* d_in, const int* in_sizes, int n_in,
                              void* d_out, int out_size, void* d_ws, size_t ws_size,
                              hipStream_t stream) {
    const float* x    = (const float*)d_in[0];
    const float* adj  = (const float*)d_in[1];
    const float* W1   = (const float*)d_in[2];
    const float* a1   = (const float*)d_in[3];
    const float* W2   = (const float*)d_in[4];
    const float* a2   = (const float*)d_in[5];
    const float* W3   = (const float*)d_in[6];
    const float* a3   = (const float*)d_in[7];
    const float* Wg0  = (const float*)d_in[8];
    const float* bg0  = (const float*)d_in[9];
    const float* Wg1  = (const float*)d_in[10];
    const float* bg1  = (const float*)d_in[11];
    const float* Wr   = (const float*)d_in[12];
    const float* br   = (const float*)d_in[13];
    const int*   rel  = (const int*)d_in[14];
    float* out = (float*)d_out;

    float* ws = (float*)d_ws;
    float* Wh       = ws;                                 // 3*4096*64
    float* sSrc     = Wh       + (size_t)3 * NN * H0;     // 3*4096
    float* sDst     = sSrc     + (size_t)3 * NN;          // 3*4096
    float* hattn    = sDst     + (size_t)3 * NN;          // 3*4096*64
    float* hprime   = hattn    + (size_t)3 * NN * H0;     // 4096*64
    float* support0 = hprime   + (size_t)NN * H0;         // 4096*64
    float* out1     = support0 + (size_t)NN * H1;         // 4096*64
    float* support1 = out1     + (size_t)NN * H1;         // 4096*32
    float* resid    = support1 + (size_t)NN * H2;         // 4096*32

    // K1: 768 waves -> 96 blocks of 256
    k_gemm_wh3<<<96, 256, 0, stream>>>(x, W1, W2, W3, Wh);
    // K2
    k_scores<<<(3 * NN + 255) / 256, 256, 0, stream>>>(Wh, a1, a2, a3, sSrc, sDst);
    // K3
    k_attn_aggregate<<<dim3(NN, 3), 256, 0, stream>>>(adj, rel, Wh, sSrc, sDst, hattn);
    // K4
    k_combine<<<(NN * H0) / 256, 256, 0, stream>>>(hattn, hprime);
    // K5: 256 waves -> 32 blocks
    k_gemm_support0<<<32, 256, 0, stream>>>(hprime, Wg0, support0);
    // K6: keep adj[rel] resident (regular temporal hints)
    k_gnn_aggregate<<<NN, 256, 0, stream>>>(adj, rel, support0, bg0, nullptr, out1, H1, 0);
    // K7: 256 waves -> 32 blocks
    k_gemm_layer2<<<32, 256, 0, stream>>>(out1, Wg1, Wr, br, support1, resid);
    // K8: last use of adj -> NT loads; final into d_out [4096,32]
    k_gnn_aggregate<<<NN, 256, 0, stream>>>(adj, rel, support1, bg1, resid, out, H2, 1);
}